// RankingLoss_24429773979806
// MI455X (gfx1250) — compile-verified
//
#include <hip/hip_runtime.h>
#include <hip/hip_bf16.h>
#include <stdint.h>

// ---------------------------------------------------------------------------
// RankingLoss on MI455X (gfx1250)
//   - Column-tiled: each block owns a 16-column slice of count/pred for all
//     2048 rows (256KB), loaded into LDS ONCE via the Tensor Data Mover,
//     then all 32 permutation passes run out of LDS. HBM traffic = one read
//     of the inputs (~328MB -> ~14us at 23.3 TB/s) instead of ~21GB naive.
//   - Per-pass permutation staged in LDS; inner loop uses float4 (B128) LDS
//     loads, 4 threads x 4 columns per row.
//   - Permutations built on-device (bitonic sort of (group, noise) keys),
//     matching the reference's double-argsort construction.
//   - Block reduction finishes with V_WMMA_F32_16X16X4_F32 (ones-vector sum).
// ---------------------------------------------------------------------------

#define N_ROWS   2048
#define N_PASS   32
#define TILE_W   16
#define BLK      1024

typedef __attribute__((ext_vector_type(4))) unsigned int v4u;
typedef __attribute__((ext_vector_type(8))) int   v8i;
typedef __attribute__((ext_vector_type(4))) int   v4i;
typedef __attribute__((ext_vector_type(2))) float v2f;
typedef __attribute__((ext_vector_type(4))) float v4f;
typedef __attribute__((ext_vector_type(8))) float v8f;

__device__ __forceinline__ unsigned hash_u32(unsigned x) {
    x ^= x >> 16; x *= 0x7feb352du;
    x ^= x >> 15; x *= 0x846ca68bu;
    x ^= x >> 16;
    return x;
}

// ---------------------------------------------------------------------------
// Kernel 0: zero the output accumulator (runs every launch; deterministic).
// ---------------------------------------------------------------------------
__global__ void rl_init_out(float* out, int n) {
    for (int i = threadIdx.x; i < n; i += blockDim.x) out[i] = 0.0f;
}

// ---------------------------------------------------------------------------
// Kernel 1: build the 32 within-group permutations.
//   order      = argsort(group)            (stable: key = group<<32 | i)
//   order_shuf = argsort(group + U(0,1))   (key = group<<32 | hash(p,i))
//   perm[order[k]] = order_shuf[k]
// One block per pass; bitonic sort of 2048 elements in LDS.
// ---------------------------------------------------------------------------
__device__ void bitonic_sort_2048(unsigned long long* key, int* val, int tid) {
    for (int k = 2; k <= N_ROWS; k <<= 1) {
        for (int j = k >> 1; j > 0; j >>= 1) {
            for (int i = tid; i < N_ROWS; i += BLK) {
                int l = i ^ j;
                if (l > i) {
                    bool up = ((i & k) == 0);
                    unsigned long long ki = key[i], kl = key[l];
                    if ((ki > kl) == up) {
                        key[i] = kl; key[l] = ki;
                        int vi = val[i]; val[i] = val[l]; val[l] = vi;
                    }
                }
            }
            __syncthreads();
        }
    }
}

__global__ __launch_bounds__(BLK)
void rl_build_perms(const int* __restrict__ groups, int* __restrict__ perms) {
    __shared__ unsigned long long key[N_ROWS];
    __shared__ int val[N_ROWS];
    __shared__ int ord[N_ROWS];
    const int p   = blockIdx.x;
    const int tid = threadIdx.x;

    // pass A: stable sort by group -> ord
    for (int i = tid; i < N_ROWS; i += BLK) {
        key[i] = ((unsigned long long)(unsigned)groups[i] << 32) | (unsigned)i;
        val[i] = i;
    }
    __syncthreads();
    bitonic_sort_2048(key, val, tid);
    for (int i = tid; i < N_ROWS; i += BLK) ord[i] = val[i];
    __syncthreads();

    // pass B: sort by (group, per-pass noise) -> order_shuf
    for (int i = tid; i < N_ROWS; i += BLK) {
        unsigned h = hash_u32((unsigned)i * 0x9e3779b9u ^ (unsigned)(p + 1) * 0x85ebca6bu);
        key[i] = ((unsigned long long)(unsigned)groups[i] << 32) | h;
        val[i] = i;
    }
    __syncthreads();
    bitonic_sort_2048(key, val, tid);

    for (int k = tid; k < N_ROWS; k += BLK)
        perms[p * N_ROWS + ord[k]] = val[k];
}

// ---------------------------------------------------------------------------
// TDM: one 2D descriptor moves a [tile_h x tile_w] fp32 tile (row stride =
// row_stride elems) from global into LDS. Tracked on TENSORcnt.
// Descriptor layout per CDNA5 ISA; 6-arg builtin.
// ---------------------------------------------------------------------------
__device__ __forceinline__ void tdm_load_tile_f32(
    const void* gaddr, unsigned lds_off,
    unsigned tile_w, unsigned tile_h,
    unsigned tensor_w, unsigned tensor_h, unsigned row_stride)
{
    unsigned long long ga = (unsigned long long)(uintptr_t)gaddr;
    v4u g0;
    g0[0] = 1u;                                              // count=1 (valid), user mode
    g0[1] = lds_off;                                         // lds_addr (bytes)
    g0[2] = (unsigned)ga;                                    // global_addr[31:0]
    g0[3] = (unsigned)((ga >> 32) & 0x01FFFFFFull)           // global_addr[56:32]
          | 0x80000000u;                                     // type=2 ("image")
    v8i g1;
    g1[0] = 0x00020000;                                      // data_size=2 (4B)
    g1[1] = (int)((tensor_w & 0xFFFFu) << 16);               // tensor_dim0[15:0]
    g1[2] = (int)((tensor_w >> 16) | ((tensor_h & 0xFFFFu) << 16)); // dim0[31:16], dim1[15:0]
    g1[3] = (int)((tensor_h >> 16) | ((tile_w & 0xFFFFu) << 16));   // dim1[31:16], tile_dim0
    g1[4] = (int)(tile_h & 0xFFFFu);                         // tile_dim1 (tile_dim2=0)
    g1[5] = (int)row_stride;                                 // tensor_dim0_stride[31:0]
    g1[6] = 0;                                               // stride0[47:32], stride1[15:0]
    g1[7] = 0;                                               // stride1[47:16]
    v4i gz4 = {0, 0, 0, 0};                                  // 2D: groups 2/3 unused
    v8i gz8 = {0, 0, 0, 0, 0, 0, 0, 0};
    __builtin_amdgcn_tensor_load_to_lds(g0, g1, gz4, gz4, gz8, 0);
}

__device__ __forceinline__ float relu_term(float ci, float cj, float pi, float pj) {
    const float y = (ci > cj) ? 1.0f : -1.0f;
    const float t = -y * (pi - pj);
    return (t > 0.0f) ? t : 0.0f;
}

// ---------------------------------------------------------------------------
// Kernel 2: main loss. grid.x = G/TILE_W blocks; each block:
//   1) TDM-loads count[:, c0:c0+16] and pred[:, c0:c0+16] into LDS (256KB)
//   2) per pass: stage perm column in LDS, accumulate relu terms with
//      float4 LDS loads (4 threads x 4 cols per row)
//   3) LDS tree reduce -> wave0 -> V_WMMA_F32_16X16X4_F32 ones-sum
//   4) global_atomic_add_f32 of the scaled block partial
// ---------------------------------------------------------------------------
__global__ __launch_bounds__(BLK)
void rl_main(const float* __restrict__ count, const float* __restrict__ pred,
             const int* __restrict__ perms, float* __restrict__ out,
             int g /*columns*/, float scale)
{
    extern __shared__ char smem[];
    float* cnt  = (float*)(smem);                    // [2048][16] = 128KB
    float* prd  = (float*)(smem + 131072);           // [2048][16] = 128KB
    float* red  = (float*)(smem + 262144);           // [1024]     =   4KB
    int*   pj_s = (int*)  (smem + 266240);           // [2048]     =   8KB

    const int tid = threadIdx.x;
    const int c0  = blockIdx.x * TILE_W;
    const int w   = (g - c0 < TILE_W) ? (g - c0) : TILE_W;

    // LDS byte offset of the dynamic segment: flat addr low 32 bits are the
    // offset within this wave's LDS allocation (CDNA5 flat->LDS mapping).
    const unsigned lds_base = (unsigned)(uintptr_t)smem;

    if (tid < 32) {  // wave 0 issues both DMA descriptors
        tdm_load_tile_f32(count + c0, lds_base,          (unsigned)w, N_ROWS,
                          (unsigned)g, N_ROWS, (unsigned)g);
        tdm_load_tile_f32(pred + c0,  lds_base + 131072, (unsigned)w, N_ROWS,
                          (unsigned)g, N_ROWS, (unsigned)g);
        __builtin_amdgcn_s_wait_tensorcnt(0);
    }
    __syncthreads();

    float acc = 0.0f;
    if (w == TILE_W) {
        // Fast path: 4 threads per row, 4 consecutive columns each (B128).
        const int q     = (tid & 3) * 4;       // column offset: 0,4,8,12
        const int rbase = tid >> 2;            // first row for this thread
        for (int p = 0; p < N_PASS; ++p) {
            const int* permp = perms + p * N_ROWS;
            for (int i = tid; i < N_ROWS; i += BLK) pj_s[i] = permp[i];
            __syncthreads();
            #pragma unroll 2
            for (int i = rbase; i < N_ROWS; i += BLK / 4) {   // 8 iterations
                const int j = pj_s[i];
                const v4f ci = *(const v4f*)&cnt[i * TILE_W + q];
                const v4f cj = *(const v4f*)&cnt[j * TILE_W + q];
                const v4f pi = *(const v4f*)&prd[i * TILE_W + q];
                const v4f pj = *(const v4f*)&prd[j * TILE_W + q];
                acc += relu_term(ci[0], cj[0], pi[0], pj[0]);
                acc += relu_term(ci[1], cj[1], pi[1], pj[1]);
                acc += relu_term(ci[2], cj[2], pi[2], pj[2]);
                acc += relu_term(ci[3], cj[3], pi[3], pj[3]);
            }
            __syncthreads();   // protect pj_s before next pass overwrites it
        }
    } else {
        // Ragged last tile: TDM packed rows at stride w; scalar loop.
        for (int p = 0; p < N_PASS; ++p) {
            const int* permp = perms + p * N_ROWS;
            for (int i = tid; i < N_ROWS; i += BLK) pj_s[i] = permp[i];
            __syncthreads();
            for (int idx = tid; idx < N_ROWS * w; idx += BLK) {
                const int i = idx / w;
                const int c = idx - i * w;
                const int j = pj_s[i];
                acc += relu_term(cnt[i * w + c], cnt[j * w + c],
                                 prd[i * w + c], prd[j * w + c]);
            }
            __syncthreads();
        }
    }

    // Block tree-reduction down to 32 lane partials in wave 0.
    red[tid] = acc;
    __syncthreads();
    for (int s = BLK / 2; s >= 32; s >>= 1) {
        if (tid < s) red[tid] += red[tid + s];
        __syncthreads();
    }

    if (tid < 32) {
        // 32 partials -> A matrix (16x4, VGPR0=partials, VGPR1=0); B = ones.
        // D[m,n] = A[m,0] + A[m,2]; summing D's 8 VGPRs then lanes {0,16}
        // yields the total of all 32 partials.
        v2f a; a[0] = red[tid]; a[1] = 0.0f;
        v2f b; b[0] = 1.0f;     b[1] = 1.0f;
        v8f cm = {};
        cm = __builtin_amdgcn_wmma_f32_16x16x4_f32(
                 false, a, false, b, (short)0, cm, false, false);
        float s = cm[0] + cm[1] + cm[2] + cm[3] + cm[4] + cm[5] + cm[6] + cm[7];
        float total = __shfl(s, 0, 32) + __shfl(s, 16, 32);
        if (tid == 0) atomicAdd(out, total * scale);
    }
}

// ---------------------------------------------------------------------------
// Host entry
// ---------------------------------------------------------------------------
extern "C" void kernel_launch(void* const* d_in, const int* in_sizes, int n_in,
                              void* d_out, int out_size, void* d_ws, size_t ws_size,
                              hipStream_t stream) {
    const float* count  = (const float*)d_in[0];
    const float* pred   = (const float*)d_in[1];
    const int*   groups = (const int*)d_in[2];
    float*       out    = (float*)d_out;
    int*         perms  = (int*)d_ws;           // 32*2048*4 = 256KB scratch

    const int n = in_sizes[2];                  // 2048
    const int g = in_sizes[0] / n;              // 20000
    (void)n_in; (void)ws_size;

    rl_init_out<<<1, 64, 0, stream>>>(out, out_size);
    rl_build_perms<<<N_PASS, BLK, 0, stream>>>(groups, perms);

    const size_t lds_bytes = 131072u * 2 + BLK * 4u + N_ROWS * 4u; // 274,432 B
    (void)hipFuncSetAttribute((const void*)rl_main,
                              hipFuncAttributeMaxDynamicSharedMemorySize,
                              (int)lds_bytes);
    const float scale = 1.0f / ((float)n * (float)g * (float)N_PASS);
    const int nblk = (g + TILE_W - 1) / TILE_W; // 1250
    rl_main<<<nblk, BLK, lds_bytes, stream>>>(count, pred, perms, out, g, scale);
}